// LSTMCell_18708877541467
// MI455X (gfx1250) — compile-verified
//
#include <hip/hip_runtime.h>

typedef float v2f __attribute__((ext_vector_type(2)));
typedef float v4f __attribute__((ext_vector_type(4)));
typedef float v8f __attribute__((ext_vector_type(8)));

// Problem constants (from reference): B=64, MAP_UNITS=512, NUM_ATTRS=64, HIDDEN=512
#define MTOT   32768   // B * MAP_UNITS (flattened GEMM rows)
#define NATTR  64
#define HID    512
#define MAPU   512

#define MT 64          // rows per workgroup
#define NT 32          // hidden cols per workgroup
#define KC 64          // K chunk staged in LDS

#define A_STRIDE 68    // LDS row stride for A tile (conflict-free frag reads)
#define B_STRIDE 40    // LDS row stride for B tiles (half-wave bank sets disjoint)

__device__ __forceinline__ float sigmoid_f(float x) { return 1.0f / (1.0f + __expf(-x)); }
__device__ __forceinline__ float tanh_f(float x)    { return 1.0f - 2.0f / (__expf(2.0f * x) + 1.0f); }

__global__ __launch_bounds__(256) void lstm_fused_wmma(
    const float* __restrict__ x,    // (MTOT, 64)
    const float* __restrict__ h,    // (MTOT, 512)
    const float* __restrict__ c,    // (MTOT, 512)
    const float* __restrict__ w_i, const float* __restrict__ u_i, const float* __restrict__ b_i,
    const float* __restrict__ w_f, const float* __restrict__ u_f, const float* __restrict__ b_f,
    const float* __restrict__ w_o, const float* __restrict__ u_o, const float* __restrict__ b_o,
    const float* __restrict__ w_g, const float* __restrict__ u_g, const float* __restrict__ b_g,
    float* __restrict__ out)        // h_new (MTOT*512) then c_new (MTOT*512)
{
    __shared__ float Alds[MT * A_STRIDE];          // 64 x 64 (stride 68)  ~17.4 KB
    __shared__ float Blds[4 * KC * B_STRIDE];      // 4 gates x 64 x 32 (stride 40) ~40 KB

    const int tid  = threadIdx.x;
    const int wave = tid >> 5;
    const int lane = tid & 31;
    const int mi   = wave >> 1;    // 0..3 : which 16-row subtile
    const int ni   = wave & 1;     // 0..1 : which 16-col subtile

    const int rowBase = blockIdx.x * MT;   // 0..32704
    const int colBase = blockIdx.y * NT;   // 0..480

    v8f acc0 = {}, acc1 = {}, acc2 = {}, acc3 = {};

    // Fragment addressing constants (per ISA 16x16x4 f32 layouts):
    //   A: lanes 0-15 hold rows M=lane, K={0,1}; lanes 16-31 rows M=lane-16, K={2,3}
    //   B: lanes 0-15 hold cols N=lane, K={0,1}; lanes 16-31 cols N=lane-16, K={2,3}
    const int rsel = lane >> 4;            // 0 or 1 (K half select)
    const int r    = lane & 15;
    const float* Abase = &Alds[(mi * 16 + r) * A_STRIDE + 2 * rsel];
    const int nloc = ni * 16 + r;          // local column 0..31

    // K = [ x@W : 64 ] ++ [ h@U : 512 ]  -> 9 chunks of 64
    for (int ch = 0; ch < 9; ++ch) {
        // ---- stage A chunk: 64 rows x 64 K-values (1024 float4, 4 per thread) ----
#pragma unroll
        for (int t = 0; t < 4; ++t) {
            const int idx = tid + t * 256;
            const int row = idx >> 4;
            const int c4  = idx & 15;
            const float* src = (ch == 0)
                ? (x + (size_t)(rowBase + row) * NATTR + c4 * 4)
                : (h + (size_t)(rowBase + row) * HID + (ch - 1) * KC + c4 * 4);
            *(v4f*)&Alds[row * A_STRIDE + c4 * 4] = *(const v4f*)src;
        }
        // ---- stage B chunk: 4 gates x 64 K x 32 N (2048 float4, 8 per thread) ----
#pragma unroll
        for (int t = 0; t < 8; ++t) {
            const int gate = t >> 1;                 // compile-time per unrolled t
            const int rem  = ((t & 1) << 8) + tid;   // 0..511 within gate
            const int row  = rem >> 3;               // K row 0..63
            const int c4   = rem & 7;                // float4 col 0..7
            const float* wp;
            if (ch == 0) {
                wp = (gate == 0) ? w_i : (gate == 1) ? w_f : (gate == 2) ? w_o : w_g;
            } else {
                wp = (gate == 0) ? u_i : (gate == 1) ? u_f : (gate == 2) ? u_o : u_g;
            }
            const int krow = (ch == 0) ? row : (ch - 1) * KC + row;  // both W and U have row stride 512
            *(v4f*)&Blds[(gate * KC + row) * B_STRIDE + c4 * 4] =
                *(const v4f*)(wp + (size_t)krow * HID + colBase + c4 * 4);
        }
        __syncthreads();

        // ---- 16 WMMA K-steps of 4, per gate ----
#pragma unroll
        for (int kk = 0; kk < 16; ++kk) {
            const int k0 = kk * 4;
            v2f a = *(const v2f*)(Abase + k0);
            const int kb = (k0 + 2 * rsel) * B_STRIDE + nloc;

            v2f b0; b0.x = Blds[0 * KC * B_STRIDE + kb]; b0.y = Blds[0 * KC * B_STRIDE + kb + B_STRIDE];
            acc0 = __builtin_amdgcn_wmma_f32_16x16x4_f32(false, a, false, b0, (short)0, acc0, false, false);
            v2f b1; b1.x = Blds[1 * KC * B_STRIDE + kb]; b1.y = Blds[1 * KC * B_STRIDE + kb + B_STRIDE];
            acc1 = __builtin_amdgcn_wmma_f32_16x16x4_f32(false, a, false, b1, (short)0, acc1, false, false);
            v2f b2; b2.x = Blds[2 * KC * B_STRIDE + kb]; b2.y = Blds[2 * KC * B_STRIDE + kb + B_STRIDE];
            acc2 = __builtin_amdgcn_wmma_f32_16x16x4_f32(false, a, false, b2, (short)0, acc2, false, false);
            v2f b3; b3.x = Blds[3 * KC * B_STRIDE + kb]; b3.y = Blds[3 * KC * B_STRIDE + kb + B_STRIDE];
            acc3 = __builtin_amdgcn_wmma_f32_16x16x4_f32(false, a, false, b3, (short)0, acc3, false, false);
        }
        __syncthreads();
    }

    // ---- fused LSTM epilogue ----
    // C/D layout: VGPR v -> row M = v + 8*(lane>=16); col N = lane&15
    const int gn    = colBase + nloc - ni * 16 + ni * 16; // = colBase + nloc
    const int rhalf = rsel * 8;
    float* out_h = out;
    float* out_c = out + (size_t)MTOT * HID;

#pragma unroll
    for (int v = 0; v < 8; ++v) {
        const int gm = rowBase + mi * 16 + rhalf + v;
        const int mu = gm & (MAPU - 1);                 // map-unit index for bias
        const size_t bidx = (size_t)mu * HID + gn;
        const float pi = acc0[v] + b_i[bidx];
        const float pf = acc1[v] + b_f[bidx];
        const float po = acc2[v] + b_o[bidx];
        const float pg = acc3[v] + b_g[bidx];
        const float ig = sigmoid_f(pi);
        const float fg = sigmoid_f(pf);
        const float og = sigmoid_f(po);
        const float gg = tanh_f(pg);
        const size_t oidx = (size_t)gm * HID + gn;
        const float cn = fg * c[oidx] + ig * gg;
        out_c[oidx] = cn;
        out_h[oidx] = og * tanh_f(cn);
    }
}

extern "C" void kernel_launch(void* const* d_in, const int* in_sizes, int n_in,
                              void* d_out, int out_size, void* d_ws, size_t ws_size,
                              hipStream_t stream) {
    const float* x   = (const float*)d_in[0];
    const float* h   = (const float*)d_in[1];
    const float* c   = (const float*)d_in[2];
    const float* w_i = (const float*)d_in[3];
    const float* u_i = (const float*)d_in[4];
    const float* b_i = (const float*)d_in[5];
    const float* w_f = (const float*)d_in[6];
    const float* u_f = (const float*)d_in[7];
    const float* b_f = (const float*)d_in[8];
    const float* w_o = (const float*)d_in[9];
    const float* u_o = (const float*)d_in[10];
    const float* b_o = (const float*)d_in[11];
    const float* w_g = (const float*)d_in[12];
    const float* u_g = (const float*)d_in[13];
    const float* b_g = (const float*)d_in[14];

    dim3 grid(MTOT / MT, HID / NT);   // 512 x 16
    lstm_fused_wmma<<<grid, 256, 0, stream>>>(
        x, h, c,
        w_i, u_i, b_i,
        w_f, u_f, b_f,
        w_o, u_o, b_o,
        w_g, u_g, b_g,
        (float*)d_out);
}